// RetraceLossRecursive_34806414967036
// MI455X (gfx1250) — compile-verified
//
#include <hip/hip_runtime.h>

typedef __attribute__((ext_vector_type(2))) float v2f;
typedef __attribute__((ext_vector_type(8))) float v8f;

#define B_DIM 256
#define T_DIM 257
#define A_DIM 64

__device__ __forceinline__ float huber(float d) {
    float ad = fabsf(d);
    return (ad < 1.0f) ? 0.5f * d * d : ad - 0.5f;
}

// One thread per (b, a). Backward time scan, fully coalesced streaming.
// 64 blocks x 256 threads = 16384 threads = 512 wave32s.
__global__ __launch_bounds__(256) void retrace_main(
    const float* __restrict__ savs,   // (B,T,A) state_trajectory_action_values
    const float* __restrict__ tQ,     // (B,T,A) target_state_trajectory_action_values
    const float* __restrict__ tV,     // (B,T,A) target_expected_state_values
    const float* __restrict__ rew,    // (B,T,A) rewards
    const float* __restrict__ olp,    // (B,T)   original_log_trajectory_action_probs
    const float* __restrict__ tlp,    // (B,T,A) target_log_trajectory_task_action_probs
    float* __restrict__ partial)      // (gridDim.x) block partial sums
{
    const int tid = threadIdx.x;
    const int b   = blockIdx.x * 4 + (tid >> 6);   // 4 batch rows per block
    const int a   = tid & 63;
    const long base  = ((long)b * T_DIM) * A_DIM + a;   // [b,0,a]
    const long obase = (long)b * T_DIM;

    const float GAMMA = 0.99f;

    // carry = savs[b, T-1, a]
    float carry = __builtin_nontemporal_load(&savs[base + 256L * A_DIM]);
    // t = 255 term: d = savs[b,255,a] - init
    float acc = huber(__builtin_nontemporal_load(&savs[base + 255L * A_DIM]) - carry);

    // Deep unroll: loads of future steps are independent of the serial carry
    // chain, so the compiler can keep ~40 NT loads in flight per wave.
    #pragma unroll 8
    for (int j = 255; j >= 1; --j) {
        const long o1 = base + (long)(j + 1) * A_DIM;
        float r   = __builtin_nontemporal_load(&rew[base + (long)j * A_DIM]);
        float v   = __builtin_nontemporal_load(&tV[o1]);
        float tq  = __builtin_nontemporal_load(&tQ[o1]);
        float lt  = __builtin_nontemporal_load(&tlp[o1]);
        float liw = lt - olp[obase + j + 1];          // olp reused across lanes: keep temporal
        float c   = __expf(fminf(liw, 0.0f));
        // R = r + GAMMA * (v + c*(carry - tq))
        float R   = fmaf(GAMMA, fmaf(c, carry - tq, v), r);
        float q   = __builtin_nontemporal_load(&savs[base + (long)(j - 1) * A_DIM]);
        acc += huber(q - R);
        carry = R;
    }

    // deterministic block reduction
    __shared__ float smem[256];
    smem[tid] = acc;
    __syncthreads();
    #pragma unroll
    for (int s = 128; s > 0; s >>= 1) {
        if (tid < s) smem[tid] += smem[tid + s];
        __syncthreads();
    }
    if (tid == 0) partial[blockIdx.x] = smem[0];
}

// Single wave32: reduce 64 partials to the final mean.
// Uses V_WMMA_F32_16X16X4_F32 with B = ones to perform the cross-lane sum
// at full f32 precision (A-tile 16x4: lane m holds A[m][0], lane 16+m holds
// A[m][2]; zeros elsewhere; D row-sums collapse the tile).
__global__ __launch_bounds__(32) void retrace_reduce(
    const float* __restrict__ partial, float* __restrict__ out)
{
    const int lane = threadIdx.x;                 // 0..31, no divergence
    float vsum = partial[lane] + partial[lane + 32];

    v2f amat; amat.x = vsum; amat.y = 0.0f;
    v2f bmat; bmat.x = 1.0f; bmat.y = 1.0f;
    v8f cacc = {};
    cacc = __builtin_amdgcn_wmma_f32_16x16x4_f32(
        /*neg_a=*/false, amat, /*neg_b=*/false, bmat,
        /*c_mod=*/(short)0, cacc, /*reuse_a=*/false, /*reuse_b=*/false);

    // lanes 0-15 hold D[m=0..7][n], lanes 16-31 hold D[m=8..15][n]
    float s = cacc[0] + cacc[1] + cacc[2] + cacc[3] +
              cacc[4] + cacc[5] + cacc[6] + cacc[7];
    s += __shfl_xor(s, 16, 32);                   // total of all 32 lane values

    const float INV_N = 1.0f / (256.0f * 256.0f * 64.0f);
    if (lane == 0) out[0] = s * INV_N;
}

extern "C" void kernel_launch(void* const* d_in, const int* in_sizes, int n_in,
                              void* d_out, int out_size, void* d_ws, size_t ws_size,
                              hipStream_t stream) {
    const float* savs = (const float*)d_in[0];
    const float* tQ   = (const float*)d_in[1];
    const float* tV   = (const float*)d_in[2];
    const float* rew  = (const float*)d_in[3];
    const float* olp  = (const float*)d_in[4];
    const float* tlp  = (const float*)d_in[5];
    float* partial = (float*)d_ws;   // 64 floats

    retrace_main<<<64, 256, 0, stream>>>(savs, tQ, tV, rew, olp, tlp, partial);
    retrace_reduce<<<1, 32, 0, stream>>>(partial, (float*)d_out);
}